// InfoMaxTrainer_75977971466801
// MI455X (gfx1250) — compile-verified
//
#include <hip/hip_runtime.h>
#include <hip/hip_bf16.h>
#include <math.h>

// ---------------------------------------------------------------------------
// InfoMax GNN forward for MI455X (gfx1250, wave32).
//  - Edge-encoder matmul algebraically hoisted: segment_sum(ea@W) = (segment_sum ea)@W
//    -> per-node attr_sum[N,10] computed once (self-loop one-hot folded into init).
//  - h-gather/scatter via hw f32 atomics (L2-resident: h=25.6MB, agg=51MB << 192MB L2).
//  - Dense GEMMs on V_WMMA_F32_16X16X4_F32, 4 N-tiles per wave (A-fragment reuse,
//    4 independent accumulators -> back-to-back WMMA issue, 4x less A traffic).
// ---------------------------------------------------------------------------

typedef __attribute__((ext_vector_type(2))) float v2f;
typedef __attribute__((ext_vector_type(8))) float v8f;

#define H_DIM 128
#define G_CNT 256
#define BLK 256

__device__ __forceinline__ void atomAddF(float* p, float v) {
  unsafeAtomicAdd(p, v);   // global_atomic_add_f32 (no CAS loop)
}

// ----------------------------- utility ------------------------------------
__global__ void zero_kernel(float* __restrict__ p, long long n) {
  long long t = (long long)blockIdx.x * blockDim.x + threadIdx.x;
  if (t < n) p[t] = 0.0f;
}

// h[n,:] = node_emb[x_ids[n],:]
__global__ void embed_kernel(const int* __restrict__ ids, const float* __restrict__ emb,
                             float* __restrict__ h, long long total) {
  long long t = (long long)blockIdx.x * blockDim.x + threadIdx.x;
  if (t >= total) return;
  long long n = t / H_DIM;
  int c = (int)(t % H_DIM);
  h[t] = emb[(long long)ids[n] * H_DIM + c];
}

// attr_sum init with self-loop contribution: one-hot at idx 7, count (idx 9) = 1
__global__ void attr_init_kernel(float* __restrict__ attr, long long total) {
  long long t = (long long)blockIdx.x * blockDim.x + threadIdx.x;
  if (t >= total) return;
  int j = (int)(t % 10);
  attr[t] = (j == 7 || j == 9) ? 1.0f : 0.0f;
}

// attr_sum[src[e]] += {edge_attr[e,0..8], 1}
__global__ void attr_accum_kernel(const int* __restrict__ src, const float* __restrict__ ea,
                                  float* __restrict__ attr, long long Ecnt) {
  long long e = (long long)blockIdx.x * blockDim.x + threadIdx.x;
  if (e >= Ecnt) return;
  int s = src[e];
  float* ap = attr + (long long)s * 10;
  const float* ep = ea + e * 9;
#pragma unroll
  for (int j = 0; j < 9; ++j) atomAddF(ap + j, ep[j]);
  atomAddF(ap + 9, 1.0f);
}

// agg[n, :H] = h[n,:] (self-loop gather term);
// agg[n, H:] = attr_sum[n,0:9] @ enc_w + attr_sum[n,9] * enc_b   (hoisted edge encoder)
__global__ void agg_init_kernel(const float* __restrict__ h, const float* __restrict__ attr,
                                const float* __restrict__ encw, const float* __restrict__ encb,
                                float* __restrict__ agg, long long total) {
  long long t = (long long)blockIdx.x * blockDim.x + threadIdx.x;
  if (t >= total) return;
  long long n = t / (2 * H_DIM);
  int c = (int)(t % (2 * H_DIM));
  if (c < H_DIM) {
    agg[t] = h[n * H_DIM + c];
  } else {
    int cc = c - H_DIM;
    const float* ap = attr + n * 10;
    float s = ap[9] * encb[cc];
#pragma unroll
    for (int j = 0; j < 9; ++j) s += ap[j] * encw[j * H_DIM + cc];
    agg[t] = s;
  }
}

// agg[src[e], :H] += h[dst[e], :]  — one wave per edge, float4 per lane (H=128)
__global__ void scatter_edges_kernel(const int* __restrict__ src, const int* __restrict__ dst,
                                     const float* __restrict__ h, float* __restrict__ agg,
                                     long long Ecnt) {
  long long t = (long long)blockIdx.x * blockDim.x + threadIdx.x;
  long long e = t >> 5;
  int lane = (int)(t & 31);
  if (e >= Ecnt) return;
  int s = src[e], d = dst[e];
  const float4 v = ((const float4*)(h + (long long)d * H_DIM))[lane];
  float* ap = agg + (long long)s * (2 * H_DIM) + lane * 4;
  atomAddF(ap + 0, v.x);
  atomAddF(ap + 1, v.y);
  atomAddF(ap + 2, v.z);
  atomAddF(ap + 3, v.w);
}

// ------------------------- f32 WMMA GEMM ----------------------------------
// D[M,N] = A[M,K] @ B[K,N] + bias; optional output relu.
// One wave computes a 16x64 strip: 4 N-tiles, 4 independent accumulators.
// K stepped by 4 via V_WMMA_F32_16X16X4_F32; A fragment loaded once per step.
// A 16x4 layout : VGPR j, lane-half hf -> (row=lane&15, col=k + j + 2*hf)
// B 4x16 layout : VGPR j, lane-half hf -> (row=k + j + 2*hf, col=lane&15)
// C/D 16x16     : VGPR r, lane-half hf -> (row=r + 8*hf,      col=lane&15)
__global__ void wmma_gemm_f32_kernel(const float* __restrict__ A, const float* __restrict__ B,
                                     const float* __restrict__ bias, float* __restrict__ D,
                                     int M, int K, int Ncol, int relu) {
  int wave = (int)(((long long)blockIdx.x * blockDim.x + threadIdx.x) >> 5);
  int lane = threadIdx.x & 31;
  int groupsN = Ncol >> 6;                   // groups of 4 x (16-wide) tiles
  int groups = (M >> 4) * groupsN;
  if (wave >= groups) return;                // wave-uniform: EXEC stays all-1s
  int tm = wave / groupsN;
  int tg = wave % groupsN;
  int hf = lane >> 4;
  int l16 = lane & 15;

  const float* Arow = A + (size_t)(tm * 16 + l16) * K;
  const float* B0   = B + (size_t)(tg * 64 + l16);

  v8f c0 = {}, c1 = {}, c2 = {}, c3 = {};
#pragma unroll 4
  for (int k = 0; k < K; k += 4) {
    v2f a = *(const v2f*)(Arow + k + 2 * hf);          // 8B-aligned contiguous pair
    const float* brow0 = B0 + (size_t)(k + 2 * hf) * Ncol;
    const float* brow1 = brow0 + Ncol;
    v2f b0, b1, b2, b3;
    b0.x = brow0[0];  b0.y = brow1[0];
    b1.x = brow0[16]; b1.y = brow1[16];
    b2.x = brow0[32]; b2.y = brow1[32];
    b3.x = brow0[48]; b3.y = brow1[48];
    c0 = __builtin_amdgcn_wmma_f32_16x16x4_f32(false, a, false, b0, (short)0, c0, false, false);
    c1 = __builtin_amdgcn_wmma_f32_16x16x4_f32(false, a, false, b1, (short)0, c1, false, false);
    c2 = __builtin_amdgcn_wmma_f32_16x16x4_f32(false, a, false, b2, (short)0, c2, false, false);
    c3 = __builtin_amdgcn_wmma_f32_16x16x4_f32(false, a, false, b3, (short)0, c3, false, false);
  }

  v8f acc[4] = {c0, c1, c2, c3};
#pragma unroll
  for (int g = 0; g < 4; ++g) {
    int col = tg * 64 + g * 16 + l16;
    float bv = bias ? bias[col] : 0.0f;
#pragma unroll
    for (int r = 0; r < 8; ++r) {
      float v = acc[g][r] + bv;
      if (relu) v = fmaxf(v, 0.0f);
      D[(size_t)(tm * 16 + r + 8 * hf) * Ncol + col] = v;
    }
  }
}

// ------------------------- BatchNorm --------------------------------------
// per-column sum / sumsq over N rows (blockDim.x == 2H columns)
__global__ void bn_stats_kernel(const float* __restrict__ z, float* __restrict__ bns,
                                int Nrows) {
  int c = threadIdx.x;                // 0..2H-1
  float s = 0.0f, q = 0.0f;
  for (int n = blockIdx.x; n < Nrows; n += gridDim.x) {
    float v = z[(size_t)n * (2 * H_DIM) + c];
    s += v;
    q += v * v;
  }
  atomAddF(&bns[c], s);
  atomAddF(&bns[2 * H_DIM + c], q);
}

// z = relu( (z - mu) * rsqrt(var+eps) * g + b )   (relu folded: reference does relu(z)@w2)
__global__ void bn_apply_kernel(float* __restrict__ z, const float* __restrict__ bns,
                                const float* __restrict__ g, const float* __restrict__ b,
                                long long total, int Nrows) {
  long long t = (long long)blockIdx.x * blockDim.x + threadIdx.x;
  if (t >= total) return;
  int c = (int)(t % (2 * H_DIM));
  float inv_n = 1.0f / (float)Nrows;
  float mu = bns[c] * inv_n;
  float var = bns[2 * H_DIM + c] * inv_n - mu * mu;
  float v = (z[t] - mu) * rsqrtf(var + 1e-5f) * g[c] + b[c];
  z[t] = fmaxf(v, 0.0f);
}

// ------------------------- InfoMax head -----------------------------------
__global__ void graph_pool_kernel(const float* __restrict__ h, const int* __restrict__ batch,
                                  float* __restrict__ gsum, float* __restrict__ gcnt,
                                  long long total) {
  long long t = (long long)blockIdx.x * blockDim.x + threadIdx.x;
  if (t >= total) return;
  long long n = t / H_DIM;
  int c = (int)(t % H_DIM);
  int g = batch[n];
  atomAddF(&gsum[(long long)g * H_DIM + c], h[t]);
  if (c == 0) atomAddF(&gcnt[g], 1.0f);
}

__global__ void summary_kernel(const float* __restrict__ gsum, const float* __restrict__ gcnt,
                               float* __restrict__ summ, long long total) {
  long long t = (long long)blockIdx.x * blockDim.x + threadIdx.x;
  if (t >= total) return;
  long long g = t / H_DIM;
  float m = gsum[t] / fmaxf(gcnt[g], 1.0f);
  summ[t] = 1.0f / (1.0f + expf(-m));
}

// out[n] = <h[n], sdisc[batch[n]]>, out[N+n] = <h[n], sdisc[(batch[n]+1)%G]>
__global__ void head_out_kernel(const float* __restrict__ h, const float* __restrict__ sdisc,
                                const int* __restrict__ batch, float* __restrict__ out,
                                int Nn) {
  long long t = (long long)blockIdx.x * blockDim.x + threadIdx.x;
  long long n = t >> 5;
  int lane = (int)(t & 31);
  if (n >= Nn) return;
  int g = batch[n];
  float4 hv = ((const float4*)(h + n * H_DIM))[lane];
  float4 pv = ((const float4*)(sdisc + (long long)g * H_DIM))[lane];
  float4 nv = ((const float4*)(sdisc + (long long)((g + 1) % G_CNT) * H_DIM))[lane];
  float p = hv.x * pv.x + hv.y * pv.y + hv.z * pv.z + hv.w * pv.w;
  float q = hv.x * nv.x + hv.y * nv.y + hv.z * nv.z + hv.w * nv.w;
#pragma unroll
  for (int off = 16; off > 0; off >>= 1) {
    p += __shfl_xor(p, off, 32);
    q += __shfl_xor(q, off, 32);
  }
  if (lane == 0) {
    out[n] = p;
    out[(long long)Nn + n] = q;
  }
}

// ---------------------------------------------------------------------------
static inline int cdivh(long long a, long long b) { return (int)((a + b - 1) / b); }

extern "C" void kernel_launch(void* const* d_in, const int* in_sizes, int n_in,
                              void* d_out, int out_size, void* d_ws, size_t ws_size,
                              hipStream_t stream) {
  const int*   x_ids = (const int*)d_in[0];
  const int*   eidx  = (const int*)d_in[1];
  const int*   batch = (const int*)d_in[2];
  const float* eattr = (const float*)d_in[3];
  const float* nemb  = (const float*)d_in[4];
  const float* encw  = (const float*)d_in[5];
  const float* encb  = (const float*)d_in[6];
  const float* w1    = (const float*)d_in[7];
  const float* b1    = (const float*)d_in[8];
  const float* bng   = (const float*)d_in[9];
  const float* bnb   = (const float*)d_in[10];
  const float* w2    = (const float*)d_in[11];
  const float* b2    = (const float*)d_in[12];
  const float* discw = (const float*)d_in[13];

  const int  N    = in_sizes[0];
  const long long E = (long long)in_sizes[1] / 2;
  const int  H    = H_DIM;
  const int  twoH = 2 * H_DIM;
  const int  G    = G_CNT;
  const int  L    = in_sizes[6] / H;   // enc_b is [L, H]

  const int* src = eidx;       // edge_index[0]  (scatter segments)
  const int* dst = eidx + E;   // edge_index[1]  (gather indices)

  float* ws = (float*)d_ws;
  size_t o = 0;
  float* h    = ws + o; o += (size_t)N * H;
  float* agg  = ws + o; o += (size_t)N * twoH;
  float* z    = ws + o; o += (size_t)N * twoH;
  float* attr = ws + o; o += (size_t)N * 10;
  float* bns  = ws + o; o += (size_t)2 * twoH;
  float* gsum = ws + o; o += (size_t)G * H;
  float* gcnt = ws + o; o += (size_t)G;
  float* summ = ws + o; o += (size_t)G * H;
  float* sd   = ws + o; o += (size_t)G * H;
  float* out  = (float*)d_out;

  // node embedding
  {
    long long tot = (long long)N * H;
    embed_kernel<<<cdivh(tot, BLK), BLK, 0, stream>>>(x_ids, nemb, h, tot);
  }
  // per-node attr sums (layer-invariant; self-loop folded into init)
  {
    long long tot = (long long)N * 10;
    attr_init_kernel<<<cdivh(tot, BLK), BLK, 0, stream>>>(attr, tot);
    attr_accum_kernel<<<cdivh(E, BLK), BLK, 0, stream>>>(src, eattr, attr, E);
  }

  for (int i = 0; i < L; ++i) {
    const float* encw_i = encw + (size_t)i * 9 * H;
    const float* encb_i = encb + (size_t)i * H;
    const float* w1_i   = w1   + (size_t)i * twoH * twoH;
    const float* b1_i   = b1   + (size_t)i * twoH;
    const float* bng_i  = bng  + (size_t)i * twoH;
    const float* bnb_i  = bnb  + (size_t)i * twoH;
    const float* w2_i   = w2   + (size_t)i * twoH * H;
    const float* b2_i   = b2   + (size_t)i * H;

    // agg init: self-loop h + hoisted edge-encoder term
    {
      long long tot = (long long)N * twoH;
      agg_init_kernel<<<cdivh(tot, BLK), BLK, 0, stream>>>(h, attr, encw_i, encb_i, agg, tot);
    }
    // neighbor gather-scatter (one wave per edge)
    {
      long long tot = E * 32;
      scatter_edges_kernel<<<cdivh(tot, BLK), BLK, 0, stream>>>(src, dst, h, agg, E);
    }
    // z = agg @ w1 + b1           [N,2H]x[2H,2H]  (WMMA f32, 16x64 strips)
    {
      long long waves = (long long)(N / 16) * (twoH / 64);
      wmma_gemm_f32_kernel<<<cdivh(waves * 32, BLK), BLK, 0, stream>>>(
          agg, w1_i, b1_i, z, N, twoH, twoH, 0);
    }
    // BatchNorm (batch statistics) + relu, in place
    {
      zero_kernel<<<cdivh(2 * twoH, BLK), BLK, 0, stream>>>(bns, 2 * twoH);
      bn_stats_kernel<<<256, twoH, 0, stream>>>(z, bns, N);
      long long tot = (long long)N * twoH;
      bn_apply_kernel<<<cdivh(tot, BLK), BLK, 0, stream>>>(z, bns, bng_i, bnb_i, tot, N);
    }
    // h = relu?( z @ w2 + b2 )    [N,2H]x[2H,H]  (WMMA f32, 16x64 strips)
    {
      long long waves = (long long)(N / 16) * (H / 64);
      wmma_gemm_f32_kernel<<<cdivh(waves * 32, BLK), BLK, 0, stream>>>(
          z, w2_i, b2_i, h, N, twoH, H, (i < L - 1) ? 1 : 0);
    }
  }

  // InfoMax head
  {
    long long tot = (long long)G * H + G;
    zero_kernel<<<cdivh(tot, BLK), BLK, 0, stream>>>(gsum, tot);  // gsum then gcnt (contiguous)
    long long nh = (long long)N * H;
    graph_pool_kernel<<<cdivh(nh, BLK), BLK, 0, stream>>>(h, batch, gsum, gcnt, nh);
    long long gh = (long long)G * H;
    summary_kernel<<<cdivh(gh, BLK), BLK, 0, stream>>>(gsum, gcnt, summ, gh);
    // sd = summary @ disc_w       [G,H]x[H,H]  (WMMA f32, 16x64 strips)
    long long waves = (long long)(G / 16) * (H / 64);
    wmma_gemm_f32_kernel<<<cdivh(waves * 32, BLK), BLK, 0, stream>>>(
        summ, discw, (const float*)nullptr, sd, G, H, H, 0);
    // pos/neg discriminator dots (neg summary = row-shifted -> shift the graph index)
    long long tot2 = (long long)N * 32;
    head_out_kernel<<<cdivh(tot2, BLK), BLK, 0, stream>>>(h, sd, batch, out, N);
  }
  (void)n_in; (void)out_size; (void)ws_size;
}